// SWAP_73813307949722
// MI455X (gfx1250) — compile-verified
//
#include <hip/hip_runtime.h>
#include <cstdint>
#include <utility>

// SWAP gate on 12 qubits (DIM=2, WIRES=12, C=0, T=1), D=4096, BATCH=1024.
// U is a permutation matrix that swaps bits 11 and 10 of the row index and is
// its own inverse:  out[comp][i][:] = x_comp[swap_bits_11_10(i)][:]
// Pure bandwidth problem (64 MB moved, ~2.75 us at 23.3 TB/s). Implemented as
// a permuted block copy through the CDNA5 async global<->LDS data mover, with
// 8 outstanding 512B-per-wave transfers to cover HBM latency. Row offsets are
// encoded in the async instructions' 24-bit IOFFSET immediate (which the ISA
// applies to BOTH the global and LDS address), so all 16 async ops share one
// base address.

#define N_ROWS    4096
#define N_COLS    1024            // floats per row
#define VEC_ROW   (N_COLS / 4)    // 256 float4 per row
#define ROW_BYTES 4096            // bytes per row (global stride == LDS stride)
#define RPB       8               // rows per block (32 KB staged in LDS)

#if defined(__HIP_DEVICE_COMPILE__) &&                                   \
    __has_builtin(__builtin_amdgcn_global_load_async_to_lds_b128) &&     \
    __has_builtin(__builtin_amdgcn_global_store_async_from_lds_b128)
#define USE_ASYNC_LDS 1
#else
#define USE_ASYNC_LDS 0
#endif

typedef int v4i __attribute__((ext_vector_type(4)));
typedef __attribute__((address_space(1))) v4i gv4i;   // global 16B vector
typedef __attribute__((address_space(3))) v4i lv4i;   // LDS    16B vector

__device__ __forceinline__ gv4i* as_global(const void* p) {
    return (gv4i*)(uintptr_t)p;                 // generic -> global (value preserving)
}
__device__ __forceinline__ lv4i* as_lds(const void* p) {
    return (lv4i*)(uint32_t)(uintptr_t)p;       // low 32 bits = LDS byte offset
}

#if USE_ASYNC_LDS
template <int... Rs>
__device__ __forceinline__ void issue_loads(const float4* src, float4* lds,
                                            std::integer_sequence<int, Rs...>) {
    // One shared base address; per-row offset in the IOFFSET immediate
    // (applied to both the global and LDS side by hardware).
    (__builtin_amdgcn_global_load_async_to_lds_b128(
         as_global(src), as_lds(lds), Rs * ROW_BYTES, 0), ...);
}
template <int... Rs>
__device__ __forceinline__ void issue_stores(const float4* dst, const float4* lds,
                                             std::integer_sequence<int, Rs...>) {
    (__builtin_amdgcn_global_store_async_from_lds_b128(
         as_global(dst), as_lds(lds), Rs * ROW_BYTES, 0), ...);
}
#endif

__global__ __launch_bounds__(256) void swap_perm_copy(
    const float4* __restrict__ xr,
    const float4* __restrict__ xi,
    float4* __restrict__ out)
{
    const unsigned blk   = blockIdx.x;            // 0 .. 1023
    const unsigned comp  = blk >> 9;              // 0 = real, 1 = imag
    const unsigned group = blk & 0x1FFu;          // 0 .. 511
    const unsigned row0  = group * RPB;           // aligned group of 8 rows

    // permutation swaps bits 11 and 10; low bits (incl. +0..7) are unchanged,
    // so the 8 source rows are contiguous too.
    const unsigned b11 = (row0 >> 11) & 1u;
    const unsigned b10 = (row0 >> 10) & 1u;
    const unsigned src_row0 = (row0 & 0x3FFu) | (b10 << 11) | (b11 << 10);

    const unsigned t = threadIdx.x;               // 0..255

    const float4* __restrict__ src =
        (comp ? xi : xr) + (size_t)src_row0 * VEC_ROW + t;
    float4* __restrict__ dst =
        out + ((size_t)comp * N_ROWS + row0) * VEC_ROW + t;

#if USE_ASYNC_LDS
    __shared__ float4 tile[RPB * VEC_ROW];        // 32 KB staging
    // 8 global->LDS transfers in flight per wave (8 x 512 B).
    issue_loads(src, &tile[t], std::make_integer_sequence<int, RPB>{});
#if __has_builtin(__builtin_amdgcn_s_wait_asynccnt)
    __builtin_amdgcn_s_wait_asynccnt(0);
#else
    asm volatile("s_wait_asynccnt 0" ::: "memory");
#endif
    // LDS -> global; each lane stores exactly what it loaded (no barrier).
    // S_ENDPGM's implicit wait-idle drains the outstanding stores.
    issue_stores(dst, &tile[t], std::make_integer_sequence<int, RPB>{});
#else
#pragma unroll
    for (int r = 0; r < RPB; ++r)
        dst[r * VEC_ROW] = src[r * VEC_ROW];
#endif
}

extern "C" void kernel_launch(void* const* d_in, const int* in_sizes, int n_in,
                              void* d_out, int out_size, void* d_ws, size_t ws_size,
                              hipStream_t stream) {
    // inputs: x_real [4096,1024] f32, x_imag [4096,1024] f32, U (ignored:
    // fixed, analytically-known permutation). output: [2,4096,1024] f32.
    const float4* xr  = (const float4*)d_in[0];
    const float4* xi  = (const float4*)d_in[1];
    float4*       out = (float4*)d_out;

    dim3 grid(2 * N_ROWS / RPB);   // 1024 blocks: one per (component, 8-row group)
    dim3 block(256);               // 8 waves of 32
    hipLaunchKernelGGL(swap_perm_copy, grid, block, 0, stream, xr, xi, out);
}